// Summarizer_33157147525623
// MI455X (gfx1250) — compile-verified
//
#include <hip/hip_runtime.h>
#include <stdint.h>

// Problem geometry (fixed by the reference).
#define N_SAMPLES 32768
#define STEP      256
#define N_EVENTS  16
#define BATCH     128
#define N_FRAMES  128          // N_SAMPLES / STEP: valid output frames after truncation
#define FPB       2            // frames processed per block

// Gather formulation of the frame-granular scatter-add:
//   out[b, f*256 + j] = sum_{e : idx[b,e] <= f} x[b, e, (f - idx[b,e])*256 + j]
// Each block: one batch b, two consecutive output frames (2 x 256 f32 samples).
// 64 threads (2 wave32), each thread owns 4 consecutive samples (16 B) per frame.
// Data movement uses CDNA5 async global->LDS copies (ASYNCcnt): up to 32
// independent b128 copies in flight per wave, one s_wait_asynccnt drain, then
// LDS-resident accumulation via ds_load_b128 and a single b128 store per frame.
__global__ __launch_bounds__(64)
void Summarizer_33157147525623_kernel(const float* __restrict__ x,
                                      const int* __restrict__ indices,
                                      float* __restrict__ out)
{
    __shared__ float4 smem[FPB][N_EVENTS][64];   // 32 KB: 1 KB slot per (frame, event)

    const int f0 = blockIdx.x * FPB;             // first output frame of this block
    const int b  = blockIdx.y;                   // batch index [0,128)
    const int t  = threadIdx.x;                  // [0,64): 4 samples per thread

    // Preload all 16 event indices with wide uniform loads -> one KMcnt wait
    // instead of 16 serialized s_load/s_wait chains.
    const int4* ip4 = (const int4*)(indices + b * N_EVENTS);   // 64B-aligned
    int4 iv[4];
#pragma unroll
    for (int i = 0; i < 4; ++i) iv[i] = ip4[i];
    int sidx[N_EVENTS];
#pragma unroll
    for (int i = 0; i < 4; ++i) {
        sidx[4 * i + 0] = iv[i].x;
        sidx[4 * i + 1] = iv[i].y;
        sidx[4 * i + 2] = iv[i].z;
        sidx[4 * i + 3] = iv[i].w;
    }

    // LDS byte offset of this lane's event-0/frame-0 slot. Generic pointers to
    // LDS carry the LDS byte address in their low 32 bits (flat addressing)
    // -- exactly what the async op's VDST expects.
    const uint32_t lds_lane = (uint32_t)(uintptr_t)(&smem[0][0][t]);

    const float* xb = x + (size_t)b * N_EVENTS * N_SAMPLES + 4 * t;

    unsigned mask[FPB];
#pragma unroll
    for (int q = 0; q < FPB; ++q) {
        const int f = f0 + q;
        unsigned m = 0;
#pragma unroll
        for (int e = 0; e < N_EVENTS; ++e) {
            const int k = f - sidx[e];           // source frame within event e
            if (k >= 0 && k < N_FRAMES) {        // block-uniform -> scalar branch
                m |= (1u << e);
                const float*   gp   = xb + (size_t)e * N_SAMPLES + (size_t)k * STEP;
                const uint32_t loff = lds_lane +
                    (uint32_t)((q * N_EVENTS + e) * 64 * sizeof(float4));
                // CDNA5 async copy: 16 bytes/lane, global -> LDS, no VGPR data.
                asm volatile("global_load_async_to_lds_b128 %0, %1, off"
                             :
                             : "v"(loff), "v"(gp)
                             : "memory");
            }
        }
        mask[q] = m;
    }

    // Drain this wave's outstanding async copies (ASYNCcnt -> 0). Each thread
    // reads back only LDS words its own wave's async ops wrote, so no
    // cross-wave barrier is required.
    asm volatile("s_wait_asynccnt 0x0" ::: "memory");

#pragma unroll
    for (int q = 0; q < FPB; ++q) {
        float4 acc = make_float4(0.f, 0.f, 0.f, 0.f);
#pragma unroll
        for (int e = 0; e < N_EVENTS; ++e) {
            if (mask[q] & (1u << e)) {           // uniform branch
                const float4 v = smem[q][e][t];  // ds_load_b128
                acc.x += v.x; acc.y += v.y; acc.z += v.z; acc.w += v.w;
            }
        }
        // Coalesced 16B store; frames with no events correctly store zeros
        // (also initializes the poisoned output buffer).
        float4* o = (float4*)(out + (size_t)b * N_SAMPLES + (size_t)(f0 + q) * STEP);
        o[t] = acc;
    }
}

extern "C" void kernel_launch(void* const* d_in, const int* in_sizes, int n_in,
                              void* d_out, int out_size, void* d_ws, size_t ws_size,
                              hipStream_t stream)
{
    (void)in_sizes; (void)n_in; (void)out_size; (void)d_ws; (void)ws_size;

    const float* x       = (const float*)d_in[0];   // [128, 16, 32768] f32
    const int*   indices = (const int*)d_in[1];     // [128, 16] int
    float*       out     = (float*)d_out;           // [128, 1, 32768] f32

    dim3 grid(N_FRAMES / FPB, BATCH);   // (64, 128): 2 frames per block
    dim3 block(64);                     // 2 wave32, 4 samples/thread/frame
    Summarizer_33157147525623_kernel<<<grid, block, 0, stream>>>(x, indices, out);
}